// Kernelception_blockA_12043088298095
// MI455X (gfx1250) — compile-verified
//
#include <hip/hip_runtime.h>
#include <math.h>

typedef __attribute__((ext_vector_type(2))) float v2f;
typedef __attribute__((ext_vector_type(8))) float v8f;

#define TILE_PITCH 34      // channel row pitch (conflict-free, keeps 8B align)
#define TILE_ROW   66      // padded width (-1 .. 64)
#define BT_PITCH   292     // padded K pitch for transposed packed weights

// LDS map (floats):
//   tile : [3][66][34]          = 6732   @ 0
//   Bt   : [80][292]            = 23360  @ 6732    (Bt[col][k] = W[k][col], cols 70..79 = 0)
//   w2s  : [288][10]            = 2880   @ 30092
//   psq  : [64]                 = 64     @ 32972   (||p||^2 per pixel of the row)
//   wsq  : [80]                 = 80     @ 33036   (||w||^2 per packed column)
#define SMEM_FLOATS 33116

__global__ __launch_bounds__(128) void kconv_fused(
    const float* __restrict__ x,
    const float* __restrict__ w1, const float* __restrict__ w2,
    const float* __restrict__ w3, const float* __restrict__ w4,
    const float* __restrict__ w5, const float* __restrict__ w6,
    const float* __restrict__ w7, const float* __restrict__ w8,
    float* __restrict__ out)
{
  extern __shared__ float smem[];
  float* tile = smem;
  float* Bt   = smem + 6732;
  float* w2s  = smem + 30092;
  float* psq  = smem + 32972;
  float* wsq  = smem + 33036;

  const int tid  = threadIdx.x;
  const int bx   = blockIdx.x;
  const int nimg = bx >> 6;   // image index
  const int h    = bx & 63;   // row index

  // GEMM branches in concat order: x1, x3, x4, x5, x6, x7, x8
  const float* Ws[7] = { w1, w3, w4, w5, w6, w7, w8 };

  // ---- stage input halo tile (rows h-1..h+1, cols -1..64, 32 ch), zero pad ----
  for (int idx = tid; idx < 3 * 66 * 32; idx += 128) {
    int c    = idx & 31;
    int r    = idx >> 5;
    int wcol = r % 66;
    int ki   = r / 66;
    int hg   = h + ki - 1;
    int wg   = wcol - 1;
    float v  = 0.0f;
    if (hg >= 0 && hg < 64 && wg >= 0 && wg < 64)
      v = x[(((nimg * 64 + hg) * 64) + wg) * 32 + c];
    tile[(ki * TILE_ROW + wcol) * TILE_PITCH + c] = v;
  }

  // ---- pack 7 weight matrices, transposed: Bt[col][k]; zero-pad cols 70..79 ----
  for (int idx = tid; idx < 80 * 288; idx += 128) {
    int k   = idx % 288;
    int col = idx / 288;
    float v = 0.0f;
    if (col < 70) {
      int b = col / 10;
      v = Ws[b][k * 10 + (col - b * 10)];
    }
    Bt[col * BT_PITCH + k] = v;
  }

  // ---- w2 for the L1 branch ----
  for (int idx = tid; idx < 288 * 10; idx += 128) w2s[idx] = w2[idx];

  // ---- ||w||^2 per packed column (consumed by sqdist branches) ----
  for (int col = tid; col < 80; col += 128) {
    float s = 0.0f;
    if (col < 70) {
      int b = col / 10;
      int o = col - b * 10;
      const float* w = Ws[b];
      for (int k = 0; k < 288; ++k) { float t = w[k * 10 + o]; s += t * t; }
    }
    wsq[col] = s;
  }

  __syncthreads();

  const int lane  = tid & 31;
  const int wave  = tid >> 5;
  const int w0    = wave * 16;    // first pixel (w coordinate) of this wave's tile
  const int mrow  = lane & 15;    // A-matrix row / B-matrix column for this lane
  const int khalf = lane >> 4;    // which K pair within each group of 4
  const int koff  = khalf * 2;

  v8f acc[5] = {};                // 5 column tiles of 16 (80 packed columns)
  float psqp = 0.0f;              // partial ||p||^2 (this lane's half of K)

  // ---- K loop: 72 steps of 16x16x4 f32 WMMA, A shared across 5 column tiles ----
  #pragma unroll
  for (int blk = 0; blk < 9; ++blk) {            // (ki,kj) patch offset
    const int ki = blk / 3;
    const int kj = blk - ki * 3;
    const float* arow = &tile[(ki * TILE_ROW + w0 + mrow + kj) * TILE_PITCH];
    const int kbase = blk * 32;
    #pragma unroll
    for (int cb = 0; cb < 32; cb += 4) {
      const int c = cb + koff;
      v2f a;
      a.x = arow[c];              // A[m][k]   (VGPR0: K=koff)
      a.y = arow[c + 1];          // A[m][k+1] (VGPR1: K=koff+1)
      psqp += a.x * a.x + a.y * a.y;
      const int k = kbase + c;
      #pragma unroll
      for (int t = 0; t < 5; ++t) {
        const float* bp = &Bt[(t * 16 + mrow) * BT_PITCH + k];
        v2f b;
        b.x = bp[0];              // B[k][n]
        b.y = bp[1];              // B[k+1][n]
        acc[t] = __builtin_amdgcn_wmma_f32_16x16x4_f32(
            false, a, false, b, (short)0, acc[t], false, false);
      }
    }
  }

  // ---- ||p||^2: combine K halves, publish per-pixel for D-layout consumers ----
  psqp += __shfl_xor(psqp, 16, 32);
  if (khalf == 0) psq[w0 + mrow] = psqp;
  __syncthreads();

  const size_t outrow = ((size_t)(nimg * 64 + h)) * 64 * 80;

  // ---- per-branch post-ops from the D layout: row = khalf*8+v, col = lane&15 ----
  #pragma unroll
  for (int t = 0; t < 5; ++t) {
    const int col = t * 16 + mrow;
    if (col < 70) {
      const int b  = col / 10;
      const int o  = col - b * 10;
      const int ch = (b == 0) ? o : (10 + b * 10 + o);  // x2 occupies ch 10..19
      const float wq = wsq[col];
      #pragma unroll
      for (int v = 0; v < 8; ++v) {
        const int m = khalf * 8 + v;
        const float lin = acc[t][v];
        float r;
        if (b == 0) {
          r = lin;                                        // x1: linear
        } else if (b == 1) {
          r = sqrtf(fmaxf(psq[w0 + m] - 2.0f * lin + wq, 0.0f));   // x3: L2
        } else if (b == 2) {
          r = tanhf(lin);                                 // x4: tanh
        } else if (b == 3) {
          r = expf(-fmaxf(psq[w0 + m] - 2.0f * lin + wq, 0.0f));   // x5: gaussian
        } else {
          const float u = lin + 1.0f;                     // x6..x8: (lin+1)^3
          r = u * u * u;
        }
        out[outrow + (size_t)(w0 + m) * 80 + ch] = r;
      }
    }
  }

  // ---- x2: L1-norm branch (VALU); lane = (pixel, K-half), reduce across halves ----
  float s[10] = {};
  #pragma unroll
  for (int blk = 0; blk < 9; ++blk) {
    const int ki = blk / 3;
    const int kj = blk - ki * 3;
    const float* arow = &tile[(ki * TILE_ROW + w0 + mrow + kj) * TILE_PITCH + khalf * 16];
    const float* wrow = &w2s[(blk * 32 + khalf * 16) * 10];
    #pragma unroll
    for (int c = 0; c < 16; ++c) {
      const float pv = arow[c];
      #pragma unroll
      for (int o = 0; o < 10; ++o)
        s[o] += fabsf(pv - wrow[c * 10 + o]);
    }
  }
  #pragma unroll
  for (int o = 0; o < 10; ++o) {
    const float tot = s[o] + __shfl_xor(s[o], 16, 32);
    if (khalf == 0)
      out[outrow + (size_t)(w0 + mrow) * 80 + 10 + o] = tot;
  }
}

extern "C" void kernel_launch(void* const* d_in, const int* in_sizes, int n_in,
                              void* d_out, int out_size, void* d_ws, size_t ws_size,
                              hipStream_t stream) {
  const float* x  = (const float*)d_in[0];
  const float* w1 = (const float*)d_in[1];
  const float* w2 = (const float*)d_in[2];
  const float* w3 = (const float*)d_in[3];
  const float* w4 = (const float*)d_in[4];
  const float* w5 = (const float*)d_in[5];
  const float* w6 = (const float*)d_in[6];
  const float* w7 = (const float*)d_in[7];
  const float* w8 = (const float*)d_in[8];
  float* out = (float*)d_out;

  dim3 grid(8 * 64);   // one block per (image, row)
  dim3 block(128);     // 4 waves, 16 pixels each
  size_t shmem = (size_t)SMEM_FLOATS * sizeof(float);  // ~129 KB of the 320 KB WGP LDS
  hipLaunchKernelGGL(kconv_fused, grid, block, shmem, stream,
                     x, w1, w2, w3, w4, w5, w6, w7, w8, out);
}